// NonLinearResBlock_28716151341202
// MI455X (gfx1250) — compile-verified
//
#include <hip/hip_runtime.h>
#include <math.h>

// ---- constants from the reference ----
#define R_TIA_F   10000.0f
#define VSCALE_F  9.0f
#define ALPHA_F   0.0005625f
#define INV_F     13.333333333333334f   // 1/(2*EKV_N*EKV_VT) = 1/0.075
#define CSHIFT_F  1.3333333333333333f   // EKV_VD * INV
#define BN_EPS_F  1e-5f
#define OUTSCALE_F (ALPHA_F * R_TIA_F)  // fold R_TIA into conv output

typedef __attribute__((ext_vector_type(2))) float v2f;
typedef __attribute__((ext_vector_type(8))) float v8f;

// Numerically stable softplus: max(x,0) + log(1 + exp(-|x|)).
// v_exp_f32 / v_log_f32 via fast intrinsics; exact in the regimes that matter
// (|x| can reach ~120 here; naive log(1+e^x) would overflow fp32).
__device__ __forceinline__ float softplus_f(float x) {
    float e = __expf(-fabsf(x));
    return fmaxf(x, 0.0f) + __logf(1.0f + e);
}

// One EKV summand: softplus(x)^2 - softplus(x - c)^2 with x = v*INV - t*INV.
__device__ __forceinline__ float ekv_term(float v, float tinv) {
    float x  = fmaf(v, INV_F, -tinv);
    float s1 = softplus_f(x);
    float s2 = softplus_f(x - CSHIFT_F);
    return fmaf(s1, s1, -(s2 * s2));
}

// ------------------------------------------------------------------
// EKV 3x3 conv, pad 1, stride 1, H=W=32, Cout=64, N=2.
// grid.x = N*64*4 ; block = 256 (8 waves: 8 output rows x 32 cols).
// theta row for this cout is block-uniform -> pre-scaled into LDS.
// ------------------------------------------------------------------
template<int CIN>
__global__ void __launch_bounds__(256)
ekv_conv3(const float* __restrict__ V, const float* __restrict__ theta,
          float* __restrict__ out)
{
    constexpr int H = 32, W = 32, CKK = CIN * 9;
    __shared__ float tinv[CKK];

    int b    = blockIdx.x;
    int tile = b & 3;
    int co   = (b >> 2) & 63;
    int n    = b >> 8;

    for (int i = threadIdx.x; i < CKK; i += 256)
        tinv[i] = theta[co * CKK + i] * INV_F;
    __syncthreads();

    int ow = threadIdx.x & 31;
    int oh = tile * 8 + (threadIdx.x >> 5);
    const float* Vn = V + n * CIN * H * W;

    float acc = 0.0f;
    for (int ci = 0; ci < CIN; ++ci) {
        const float* Vc = Vn + ci * H * W;
        __builtin_prefetch(Vc + H * W, 0, 0);   // global_prefetch_b8: next plane
        const float* tp = &tinv[ci * 9];
        #pragma unroll
        for (int kh = 0; kh < 3; ++kh) {
            int ih = oh + kh - 1;
            bool rowok = (unsigned)ih < (unsigned)H;
            #pragma unroll
            for (int kw = 0; kw < 3; ++kw) {
                int iw = ow + kw - 1;
                float v = (rowok && (unsigned)iw < (unsigned)W) ? Vc[ih * W + iw] : 0.0f;
                acc += ekv_term(v, tp[kh * 3 + kw]);
            }
        }
    }
    out[((n * 64 + co) * H + oh) * W + ow] = acc * OUTSCALE_F;
}

// ------------------------------------------------------------------
// Second stage: EKV 3x3 (Cin=64) over V1 + EKV 1x1 shortcut (Cin=32)
// over V_in, summed, times ALPHA*R_TIA.
// ------------------------------------------------------------------
__global__ void __launch_bounds__(256)
ekv_conv3_plus_sc(const float* __restrict__ V1, const float* __restrict__ theta2,
                  const float* __restrict__ V0, const float* __restrict__ theta_sc,
                  float* __restrict__ out)
{
    constexpr int H = 32, W = 32, CKK2 = 64 * 9, CSC = 32;
    __shared__ float tinv[CKK2 + CSC];

    int b    = blockIdx.x;
    int tile = b & 3;
    int co   = (b >> 2) & 63;
    int n    = b >> 8;

    for (int i = threadIdx.x; i < CKK2 + CSC; i += 256) {
        float t = (i < CKK2) ? theta2[co * CKK2 + i] : theta_sc[co * CSC + (i - CKK2)];
        tinv[i] = t * INV_F;
    }
    __syncthreads();

    int ow = threadIdx.x & 31;
    int oh = tile * 8 + (threadIdx.x >> 5);

    float acc = 0.0f;
    // 3x3 over V1 (64 channels)
    {
        const float* Vn = V1 + n * 64 * H * W;
        for (int ci = 0; ci < 64; ++ci) {
            const float* Vc = Vn + ci * H * W;
            __builtin_prefetch(Vc + H * W, 0, 0);
            const float* tp = &tinv[ci * 9];
            #pragma unroll
            for (int kh = 0; kh < 3; ++kh) {
                int ih = oh + kh - 1;
                bool rowok = (unsigned)ih < (unsigned)H;
                #pragma unroll
                for (int kw = 0; kw < 3; ++kw) {
                    int iw = ow + kw - 1;
                    float v = (rowok && (unsigned)iw < (unsigned)W) ? Vc[ih * W + iw] : 0.0f;
                    acc += ekv_term(v, tp[kh * 3 + kw]);
                }
            }
        }
    }
    // 1x1 shortcut over V0 (32 channels), pad 0
    {
        int pix = oh * W + ow;
        const float* Vn = V0 + n * 32 * H * W;
        for (int ci = 0; ci < 32; ++ci)
            acc += ekv_term(Vn[ci * H * W + pix], tinv[CKK2 + ci]);
    }
    out[((n * 64 + co) * H + oh) * W + ow] = acc * OUTSCALE_F;
}

// ------------------------------------------------------------------
// BatchNorm stats via the matrix pipe: per-channel sum and sum-of-squares
// as f32 GEMVs  S = X*1, Q = (X.*X)*1  using V_WMMA_F32_16X16X4_F32 with an
// all-ones B, accumulating through C->D. One wave handles 16 channels.
// x layout: [n(2), c(64), 1024]; per-channel elements = 2048.
// A layout (16x4 f32): lane l holds rows m=l%16; k = 2*(l/16) + vgpr.
// D layout: VGPR r = row r (lanes 0-15) / row r+8 (lanes 16-31); every N
// column equals the row sum since B == 1.
// ------------------------------------------------------------------
__global__ void __launch_bounds__(32)
bn_stats_wmma(const float* __restrict__ x, const float* __restrict__ gamma,
              const float* __restrict__ beta, float* __restrict__ scale,
              float* __restrict__ bias)
{
    const int NE = 2048;
    int g    = blockIdx.x;          // channel group of 16
    int lane = threadIdx.x;
    int ch   = g * 16 + (lane & 15);
    int koff = (lane >> 4) * 2;     // lanes 0-15 -> k{0,1}; lanes 16-31 -> k{2,3}

    v8f csum = {};
    v8f csq  = {};
    v2f ones; ones.x = 1.0f; ones.y = 1.0f;

    for (int base = 0; base < NE; base += 4) {
        int e0 = base + koff;
        int e1 = e0 + 1;
        int a0 = (((e0 >> 10) * 64 + ch) << 10) + (e0 & 1023);
        int a1 = (((e1 >> 10) * 64 + ch) << 10) + (e1 & 1023);
        v2f a;  a.x  = x[a0];      a.y  = x[a1];
        v2f a2; a2.x = a.x * a.x;  a2.y = a.y * a.y;
        csum = __builtin_amdgcn_wmma_f32_16x16x4_f32(false, a,  false, ones,
                                                     (short)0, csum, false, false);
        csq  = __builtin_amdgcn_wmma_f32_16x16x4_f32(false, a2, false, ones,
                                                     (short)0, csq,  false, false);
    }

    // lane 0 holds rows 0..7 in its 8 C/D VGPRs, lane 16 holds rows 8..15
    if (lane == 0 || lane == 16) {
        int cb = g * 16 + (lane >> 4) * 8;
        #pragma unroll
        for (int r = 0; r < 8; ++r) {
            float S    = csum[r];
            float Q    = csq[r];
            float mean = S * (1.0f / NE);
            float var  = Q * (1.0f / NE) - mean * mean;
            int   c    = cb + r;
            float sc   = gamma[c] * rsqrtf(var + BN_EPS_F);
            scale[c]   = sc;
            bias[c]    = beta[c] - mean * sc;
        }
    }
}

// ------------------------------------------------------------------
// y = clip(x*scale[c] + bias[c], 0, VSCALE), channel = (i>>10) & 63
// ------------------------------------------------------------------
__global__ void __launch_bounds__(256)
norm_clip(const float* __restrict__ x, const float* __restrict__ scale,
          const float* __restrict__ bias, float* __restrict__ y, int total)
{
    int i = blockIdx.x * 256 + threadIdx.x;
    if (i < total) {
        int c = (i >> 10) & 63;
        float v = fmaf(x[i], scale[c], bias[c]);
        y[i] = fminf(fmaxf(v, 0.0f), VSCALE_F);
    }
}

extern "C" void kernel_launch(void* const* d_in, const int* in_sizes, int n_in,
                              void* d_out, int out_size, void* d_ws, size_t ws_size,
                              hipStream_t stream) {
    (void)in_sizes; (void)n_in; (void)out_size; (void)ws_size;
    const float* V_in     = (const float*)d_in[0];
    const float* theta1   = (const float*)d_in[1];
    const float* theta2   = (const float*)d_in[2];
    const float* theta_sc = (const float*)d_in[3];
    const float* gamma1   = (const float*)d_in[4];
    const float* beta1    = (const float*)d_in[5];
    const float* gamma2   = (const float*)d_in[6];
    const float* beta2    = (const float*)d_in[7];
    float* out = (float*)d_out;

    float* ws  = (float*)d_ws;
    float* x1  = ws;                 // 131072
    float* V1  = ws + 131072;        // 131072
    float* x2  = ws + 262144;        // 131072
    float* sc1 = ws + 393216;        // 64
    float* b1  = sc1 + 64;
    float* sc2 = b1 + 64;
    float* b2  = sc2 + 64;

    const int NTOT = 2 * 64 * 32 * 32;   // 131072
    ekv_conv3<32><<<512, 256, 0, stream>>>(V_in, theta1, x1);
    bn_stats_wmma<<<4, 32, 0, stream>>>(x1, gamma1, beta1, sc1, b1);
    norm_clip<<<512, 256, 0, stream>>>(x1, sc1, b1, V1, NTOT);
    ekv_conv3_plus_sc<<<512, 256, 0, stream>>>(V1, theta2, V_in, theta_sc, x2);
    bn_stats_wmma<<<4, 32, 0, stream>>>(x2, gamma2, beta2, sc2, b2);
    norm_clip<<<512, 256, 0, stream>>>(x2, sc2, b2, out, NTOT);
}